// KANLayer_5231270166540
// MI455X (gfx1250) — compile-verified
//
#include <hip/hip_runtime.h>
#include <math.h>

typedef __attribute__((ext_vector_type(2))) float v2f;
typedef __attribute__((ext_vector_type(4))) float v4f;
typedef __attribute__((ext_vector_type(8))) float v8f;

#define NROWS  4096
#define NIN    64
#define NOUT   64
#define NB     11      // n_bases
#define NBP    12      // padded to multiple of 4 for K=4 WMMA steps
#define TILE_M 16      // rows per block (WMMA M)
#define TSTRIDE 20     // transpose-buffer row stride (dwords): 16B-aligned, bank-staggered

// One block: 16 rows x 64 in-features x 64 out-features.
// 128 threads = 4 waves; wave w handles out-column tile [16w, 16w+16).
__global__ __launch_bounds__(128) void kan_wmma_kernel(
    const float* __restrict__ x,      // (4096, 64)
    const float* __restrict__ grid,   // (64, 15) uniform, identical rows
    const float* __restrict__ C,      // (64, 64, 11)
    const float* __restrict__ w_base, // (64, 64)
    const float* __restrict__ w_sp,   // (64, 64)
    float* __restrict__ out,          // (4096, 64)
    float* __restrict__ post_act)     // (4096, 64, 64)
{
    // Bases laid out [i][b][row] so WMMA A-fragment reads (row = lane&15)
    // hit 16 consecutive banks; silu cache [i][row]; per-wave transpose buffer.
    __shared__ float Blds[NIN * NBP * TILE_M];       // 48 KB
    __shared__ float Slds[NIN * TILE_M];             // 4 KB
    __shared__ float Tlds[4][TILE_M * TSTRIDE];      // 5 KB

    const int tid = threadIdx.x;
    const int n0  = blockIdx.x * TILE_M;

    const float g0    = grid[0];
    const float h     = grid[1] - grid[0];
    const float inv_h = 1.0f / h;

    // ---- Phase 1: cubic B-spline bases + silu for 16 rows x 64 features ----
    // Uniform-knot closed form == Cox-de Boor on this grid.
    for (int p = 0; p < (NIN * TILE_M) / 128; ++p) {
        int pair = p * 128 + tid;
        int i    = pair >> 4;   // in-feature
        int row  = pair & 15;   // row within slab
        float xv = x[(size_t)(n0 + row) * NIN + i];

        float t = (xv - g0) * inv_h;
        int s = (int)floorf(t);
        s = (s < 3) ? 3 : s;
        s = (s > NB - 1) ? (NB - 1) : s;     // span index in [3, 10]
        float u  = t - (float)s;
        float um = 1.0f - u;
        float u2 = u * u, u3 = u2 * u;
        float w0 = um * um * um * (1.0f / 6.0f);
        float w1 = (3.0f * u3 - 6.0f * u2 + 4.0f) * (1.0f / 6.0f);
        float w2 = (-3.0f * u3 + 3.0f * u2 + 3.0f * u + 1.0f) * (1.0f / 6.0f);
        float w3 = u3 * (1.0f / 6.0f);

        #pragma unroll
        for (int b = 0; b < NBP; ++b)
            Blds[(i * NBP + b) * TILE_M + row] = 0.0f;
        int c0 = s - 3;                       // first nonzero basis, in [0, 7]
        Blds[(i * NBP + c0 + 0) * TILE_M + row] = w0;
        Blds[(i * NBP + c0 + 1) * TILE_M + row] = w1;
        Blds[(i * NBP + c0 + 2) * TILE_M + row] = w2;
        Blds[(i * NBP + c0 + 3) * TILE_M + row] = w3;

        Slds[i * TILE_M + row] = xv / (1.0f + __expf(-xv));   // silu
    }
    __syncthreads();

    // ---- Phase 2: per-feature 16x12 @ 12x16 via 3x V_WMMA_F32_16X16X4_F32 ----
    const int wave  = tid >> 5;
    const int lane  = tid & 31;
    const int jloc  = lane & 15;
    const int half  = lane >> 4;        // lanes 16..31 -> second K pair / row half
    const int jbase = wave * 16;
    const int j     = jbase + jloc;     // global out column
    const int koff  = half * 2;         // A/B fragment K offset within K=4 chunk
    const int arow  = lane & 15;        // A-fragment row
    const int mrow0 = half * 8;         // C/D layout: VGPR v holds row mrow0+v
    const float himask = (half == 0) ? 1.0f : 0.0f;  // kills b index 11 pad lane

    // store-side mapping for b128 stores out of the transpose buffer
    const int r0 = lane >> 2;           // 0..7
    const int jq = (lane & 3) * 4;      // 0,4,8,12

    float* T = &Tlds[wave][0];
    v8f acc = {0.f, 0.f, 0.f, 0.f, 0.f, 0.f, 0.f, 0.f};

    for (int i = 0; i < NIN; ++i) {
        v8f spl = {0.f, 0.f, 0.f, 0.f, 0.f, 0.f, 0.f, 0.f};
        const float* Crow = C + ((size_t)i * NOUT + j) * NB;

        #pragma unroll
        for (int kc = 0; kc < 3; ++kc) {
            int kb = kc * 4 + koff;
            v2f a, b;
            // A 16x4: lanes 0-15 rows M=0..15 with K=0,1; lanes 16-31 K=2,3
            a.x = Blds[(i * NBP + kb + 0) * TILE_M + arow];
            a.y = Blds[(i * NBP + kb + 1) * TILE_M + arow];
            // B 4x16: V0 rows K=kb (by lane half); V1 rows K=kb+1
            b.x = Crow[kb];
            if (kc < 2) {
                b.y = Crow[kb + 1];
            } else {
                // kb+1 is 9 (half 0, valid) or 11 (half 1, pad). Load the always-
                // valid clamped address, zero the pad lane arithmetically: no
                // divergent EXEC, no predicated load.
                float tval = Crow[9 + half];      // Crow[9] or Crow[10], both legal
                b.y = tval * himask;
            }
            spl = __builtin_amdgcn_wmma_f32_16x16x4_f32(
                false, a, false, b, (short)0, spl, false, false);
        }

        float ws = w_sp[(size_t)i * NOUT + j];
        float wb = w_base[(size_t)i * NOUT + j];

        // Epilogue in WMMA C/D layout; stage tile in LDS for wide stores.
        #pragma unroll
        for (int v = 0; v < 8; ++v) {
            int m = mrow0 + v;
            float val = ws * spl[v] + wb * Slds[i * TILE_M + m];
            T[m * TSTRIDE + jloc] = val;
            acc[v] += val;   // out = sum over i of post_act
        }

        // Wave-synchronous readback (LDS ops from one wave are in-order):
        // each lane owns 4 contiguous floats -> 2x global_store_b128, NT hint
        // (64 MB streaming output, never re-read on device).
        #pragma unroll
        for (int s = 0; s < 2; ++s) {
            int r = r0 + s * 8;
            v4f vv = *(const v4f*)&T[r * TSTRIDE + jq];
            float* dst = &post_act[(((size_t)(n0 + r)) * NIN + i) * NOUT + jbase + jq];
            __builtin_nontemporal_store(vv, (v4f*)dst);
        }
    }

    #pragma unroll
    for (int v = 0; v < 8; ++v) {
        int m = mrow0 + v;
        __builtin_nontemporal_store(acc[v], &out[(size_t)(n0 + m) * NOUT + j]);
    }
}

extern "C" void kernel_launch(void* const* d_in, const int* in_sizes, int n_in,
                              void* d_out, int out_size, void* d_ws, size_t ws_size,
                              hipStream_t stream) {
    (void)in_sizes; (void)n_in; (void)out_size; (void)d_ws; (void)ws_size;
    const float* x      = (const float*)d_in[0];
    const float* grid   = (const float*)d_in[1];
    const float* C      = (const float*)d_in[2];
    const float* w_base = (const float*)d_in[3];
    const float* w_sp   = (const float*)d_in[4];

    float* out      = (float*)d_out;                       // (4096, 64)
    float* post_act = out + (size_t)NROWS * NOUT;          // (4096, 64, 64)

    dim3 blocks(NROWS / TILE_M);   // 256 blocks
    kan_wmma_kernel<<<blocks, 128, 0, stream>>>(x, grid, C, w_base, w_sp,
                                                out, post_act);
}